// LaneNet_1906965479959
// MI455X (gfx1250) — compile-verified
//
#include <hip/hip_runtime.h>

typedef __attribute__((ext_vector_type(2))) float v2f;
typedef __attribute__((ext_vector_type(8))) float v8f;

#define NIMG 4
#define CCH  8
#define GN   5
#define HP   (280 * 640)   // 179200, divisible by 16

#define DELTA_V 0.2f
#define DELTA_D 1.2f

// ws layout (floats):
//   [0, NIMG*256)                : per-image 16x16 WMMA accumulation tiles
//                                  tile[n][g][c] = sum of pred over group g, col 8 = point count
//   [NIMG*256, NIMG*256+NIMG)    : n_groups per image
//   [NIMG*256+NIMG, +NIMG*GN*CCH): centers[n][g][c]
#define WS_TILES   0
#define WS_NGROUP  (NIMG * 256)
#define WS_CENTERS (NIMG * 256 + NIMG)
#define WS_FLOATS  (NIMG * 256 + NIMG + NIMG * GN * CCH)

__global__ void lanenet_zero_kernel(float* __restrict__ ws, float* __restrict__ out) {
    int i = blockIdx.x * blockDim.x + threadIdx.x;
    for (; i < WS_FLOATS; i += gridDim.x * blockDim.x) ws[i] = 0.0f;
    if (blockIdx.x == 0 && threadIdx.x < 3) out[threadIdx.x] = 0.0f;
}

// Stage 1: center sums + counts via V_WMMA_F32_16X16X4_F32, branch-free, 4x unrolled.
// Per wmma: 4 points (K=4). A[g][k] = one-hot(target==g+1), B[k][c] = pred channel c
// (c<8), B[k][8] = foreground mask (point counts land in accumulator column 8).
__global__ void lanenet_center_wmma_kernel(const float* __restrict__ preds,
                                           const int*   __restrict__ targets,
                                           float*       __restrict__ tiles) {
    const int lane          = threadIdx.x & 31;
    const int waveInBlock   = threadIdx.x >> 5;
    const int wavesPerBlock = blockDim.x >> 5;
    const int n             = blockIdx.y;
    const int waveId        = blockIdx.x * wavesPerBlock + waveInBlock;
    const int numWaves      = gridDim.x * wavesPerBlock;

    // A 16x4 f32: lanes 0-15 -> M=lane, V0=K0,V1=K1 ; lanes 16-31 -> M=lane-16, V0=K2,V1=K3
    // B 4x16 f32: lanes 0-15 -> N=lane, V0=K0,V1=K1 ; lanes 16-31 -> N=lane-16, V0=K2,V1=K3
    const int   kb       = (lane >= 16) ? 2 : 0;
    const int   m        = lane & 15;              // A row: group id
    const int   c        = lane & 15;              // B col: channel (8 = count column)
    const float chanMask = (c < CCH)  ? 1.0f : 0.0f;   // lane-constant blend masks
    const float cntMask  = (c == CCH) ? 1.0f : 0.0f;
    const int   cAddr    = c & 7;                  // clamped channel for the always-on load

    const int*   tgt = targets + n * HP;
    const float* prd = preds + (size_t)n * CCH * HP + (size_t)cAddr * HP;

    v8f acc = {};
    for (int p = waveId * 16; p < HP; p += numWaves * 16) {
#pragma unroll
        for (int u = 0; u < 4; ++u) {
            const int pp = p + u * 4 + kb;
            const int   t0 = tgt[pp];
            const int   t1 = tgt[pp + 1];
            const float q0 = prd[pp];
            const float q1 = prd[pp + 1];
            v2f a, b;
            a.x = (t0 == m + 1) ? 1.0f : 0.0f;
            a.y = (t1 == m + 1) ? 1.0f : 0.0f;
            const float m0 = (t0 > 0) ? 1.0f : 0.0f;
            const float m1 = (t1 > 0) ? 1.0f : 0.0f;
            b.x = chanMask * q0 + cntMask * m0;
            b.y = chanMask * q1 + cntMask * m1;
            acc = __builtin_amdgcn_wmma_f32_16x16x4_f32(
                /*neg_a=*/false, a, /*neg_b=*/false, b,
                /*c_mod=*/(short)0, acc, /*reuse_a=*/false, /*reuse_b=*/false);
        }
    }

    // C/D layout: VGPR r, lanes 0-15 -> M=r ; lanes 16-31 -> M=r+8 ; N = lane&15
    float* tile = tiles + n * 256;
    const int rowOff = (lane >= 16) ? 8 : 0;
    const int col    = lane & 15;
#pragma unroll
    for (int r = 0; r < 8; ++r) {
        atomicAdd(&tile[(r + rowOff) * 16 + col], acc[r]);
    }
}

// Stage 2: centers, n_groups, dist loss, reg loss (tiny; one thread per image).
__global__ void lanenet_dist_reg_kernel(const float* __restrict__ tiles,
                                        float* __restrict__ ngroups,
                                        float* __restrict__ centers,
                                        float* __restrict__ out) {
    __shared__ float s_dist[NIMG];
    __shared__ float s_reg[NIMG];
    const int n = threadIdx.x;
    if (n < NIMG) {
        const float* tile = tiles + n * 256;
        float cen[GN][CCH];
        float hp[GN];
        float ng = 0.0f;
        for (int g = 0; g < GN; ++g) {
            const float cnt = tile[g * 16 + CCH];
            hp[g] = (cnt > 0.0f) ? 1.0f : 0.0f;
            ng += hp[g];
            const float inv = 1.0f / (cnt + 1e-5f);
            for (int ch = 0; ch < CCH; ++ch) cen[g][ch] = tile[g * 16 + ch] * inv;
        }
        ngroups[n] = ng;
        for (int g = 0; g < GN; ++g)
            for (int ch = 0; ch < CCH; ++ch)
                centers[((size_t)n * GN + g) * CCH + ch] = cen[g][ch];

        // dist: rows masked by has_point[i], all columns j (diagonal included)
        float ld = 0.0f;
        for (int i = 0; i < GN; ++i) {
            if (hp[i] == 0.0f) continue;
            for (int j = 0; j < GN; ++j) {
                float ss = 0.0f;
                for (int ch = 0; ch < CCH; ++ch) {
                    const float d = cen[j][ch] - cen[i][ch];
                    ss += d * d;
                }
                const float dn = (ss > 0.0f) ? sqrtf(ss) : 0.0f;  // safe_norm
                const float v  = fmaxf(DELTA_D - dn, 0.0f);
                ld += v * v;
            }
        }
        s_dist[n] = ld / fmaxf(ng * (ng - 1.0f), 1.0f);

        float lr = 0.0f;
        for (int g = 0; g < GN; ++g) {
            float s = 0.0f;
            for (int ch = 0; ch < CCH; ++ch) s += cen[g][ch];
            lr += s * s * hp[g];
        }
        s_reg[n] = lr;
    }
    __syncthreads();
    if (threadIdx.x == 0) {
        float d = 0.0f, r = 0.0f;
        for (int k = 0; k < NIMG; ++k) { d += s_dist[k]; r += s_reg[k]; }
        out[0] = d / (float)NIMG;     // loss_dist (mean over images)
        out[2] = r * 0.001f;          // loss_reg
        // out[1] pre-zeroed; stage 3 accumulates var there
    }
}

// Stage 3: variance term, streaming pass over all points.
__global__ void lanenet_var_kernel(const float* __restrict__ preds,
                                   const int*   __restrict__ targets,
                                   const float* __restrict__ centers,
                                   const float* __restrict__ ngroups,
                                   float* __restrict__ out) {
    __shared__ float sred[256];
    const long long total  = (long long)NIMG * HP;
    const long long stride = (long long)gridDim.x * blockDim.x;
    float acc = 0.0f;
    for (long long i = (long long)blockIdx.x * blockDim.x + threadIdx.x; i < total; i += stride) {
        const int n = (int)(i / HP);
        const int p = (int)(i % HP);
        const int t = targets[(size_t)n * HP + p];
        if (t > 0) {
            const int g = t - 1;
            const float* cen = centers + ((size_t)n * GN + g) * CCH;
            const float* prd = preds + (size_t)n * CCH * HP + p;
            float ss = 0.0f;
#pragma unroll
            for (int ch = 0; ch < CCH; ++ch) {
                const float d = prd[(size_t)ch * HP] - cen[ch];
                ss += d * d;
            }
            const float dn = (ss > 0.0f) ? sqrtf(ss) : 0.0f;   // safe_norm
            const float v  = fmaxf(dn - DELTA_V, 0.0f);
            // fold: * 0.01 (var scale), / n_groups[n], / NIMG (mean)
            acc += v * v * (0.01f / (ngroups[n] * (float)NIMG));
        }
    }
    sred[threadIdx.x] = acc;
    __syncthreads();
    for (int s = blockDim.x >> 1; s > 0; s >>= 1) {
        if ((int)threadIdx.x < s) sred[threadIdx.x] += sred[threadIdx.x + s];
        __syncthreads();
    }
    if (threadIdx.x == 0) atomicAdd(&out[1], sred[0]);
}

extern "C" void kernel_launch(void* const* d_in, const int* in_sizes, int n_in,
                              void* d_out, int out_size, void* d_ws, size_t ws_size,
                              hipStream_t stream) {
    const float* preds   = (const float*)d_in[0];
    const int*   targets = (const int*)d_in[1];
    // d_in[2] = group_num (==5), compiled in as GN

    float* out     = (float*)d_out;
    float* ws      = (float*)d_ws;
    float* tiles   = ws + WS_TILES;
    float* ngroups = ws + WS_NGROUP;
    float* centers = ws + WS_CENTERS;

    lanenet_zero_kernel<<<4, 256, 0, stream>>>(ws, out);

    // 112 blocks x 8 waves per image -> 896 waves; HP/16 = 11200 unrolled chunks
    dim3 g1(112, NIMG);
    lanenet_center_wmma_kernel<<<g1, 256, 0, stream>>>(preds, targets, tiles);

    lanenet_dist_reg_kernel<<<1, 32, 0, stream>>>(tiles, ngroups, centers, out);

    lanenet_var_kernel<<<448, 256, 0, stream>>>(preds, targets, centers, ngroups, out);
}